// MyModel_61933428409198
// MI455X (gfx1250) — compile-verified
//
#include <hip/hip_runtime.h>
#include <hip/hip_bf16.h>
#include <float.h>

typedef float v2f __attribute__((ext_vector_type(2)));
typedef float v8f __attribute__((ext_vector_type(8)));

#define BATCH   32
#define NCAND   2048
#define DDIM    2048
#define KSEL    5
#define TSTRIDE 36   // 16-row LDS tile stride (floats); gcd(36,64)=4 -> conflict-free

// ---------------------------------------------------------------------------
// Kernel 1: scores[b,n] = dot(value[b,n,:], w) + bias, via V_WMMA_F32_16X16X4_F32
// A (16x4) = w[k..k+3] replicated across all 16 rows -> every D row = score row.
// B (4x16) = 16 candidate rows (N) x 4 K values, staged through LDS.
// ---------------------------------------------------------------------------
__global__ void __launch_bounds__(256)
score_kernel(const float* __restrict__ value,
             const float* __restrict__ w,
             const float* __restrict__ bias,
             float* __restrict__ scores)
{
    __shared__ __align__(16) float w_lds[DDIM];              // 8 KB
    __shared__ __align__(16) float tile[8][16 * TSTRIDE];    // 18 KB (per-wave tiles)

    const int tid  = threadIdx.x;
    const int wave = tid >> 5;
    const int lane = tid & 31;
    const int b    = blockIdx.x >> 4;        // 16 tiles of 128 rows per batch
    const int n0   = (blockIdx.x & 15) * 128 + wave * 16;

    // Stage w into LDS once per block (coalesced float4)
    {
        const float4* w4  = (const float4*)w;
        float4*       wl4 = (float4*)w_lds;
        for (int i = tid; i < DDIM / 4; i += 256) wl4[i] = w4[i];
    }
    __syncthreads();

    const float* rowbase = value + ((size_t)b * NCAND + n0) * DDIM;
    float* tl = tile[wave];

    v8f acc = {0.f, 0.f, 0.f, 0.f, 0.f, 0.f, 0.f, 0.f};
    const int koff = (lane >> 4) << 1;   // K sub-offset: lanes 0-15 -> K{0,1}, 16-31 -> K{2,3}
    const int nrow = lane & 15;          // N (candidate-row) index within tile

    for (int k0 = 0; k0 < DDIM; k0 += 32) {
        // ---- stage 16 rows x 32 K into LDS, fully coalesced (128B per row) ----
        #pragma unroll
        for (int j = 0; j < 4; ++j) {
            int f  = lane + 32 * j;          // 0..127 flat float4 index
            int r  = f >> 3;                 // row 0..15 (8 lanes cover one row)
            int c4 = (f & 7) << 2;           // col 0,4,...,28
            float4 v = *(const float4*)(rowbase + (size_t)r * DDIM + k0 + c4);
            *(float4*)(&tl[r * TSTRIDE + c4]) = v;
        }
        // same-wave LDS RAW across lanes: LDS pipe is in-order; fence the compiler + hw
        asm volatile("s_wait_dscnt 0" ::: "memory");

        // ---- 8 x WMMA f32 16x16x4, K advances by 4 each step ----
        #pragma unroll
        for (int kk = 0; kk < 32; kk += 4) {
            v2f afrag = *(const v2f*)(&w_lds[k0 + kk + koff]);                // broadcast
            v2f bfrag = *(const v2f*)(&tl[nrow * TSTRIDE + kk + koff]);       // conflict-free
            acc = __builtin_amdgcn_wmma_f32_16x16x4_f32(
                      false, afrag, false, bfrag, (short)0, acc, false, false);
        }
        // keep next iteration's LDS stores from being hoisted above these loads
        asm volatile("" ::: "memory");
    }

    // D row M=0 lives in acc[0] of lanes 0-15 (N = lane)
    if (lane < 16) {
        scores[(size_t)b * NCAND + n0 + lane] = acc[0] + bias[0];
    }
}

// ---------------------------------------------------------------------------
// Kernel 2: per-batch top-5 (descending, lower-index tie-break like lax.top_k)
// ---------------------------------------------------------------------------
__global__ void __launch_bounds__(256)
topk_kernel(const float* __restrict__ scores, int* __restrict__ sel)
{
    __shared__ float s[NCAND];     // 8 KB score row
    __shared__ float rv[256];
    __shared__ int   ri[256];

    const int b   = blockIdx.x;
    const int tid = threadIdx.x;

    for (int i = tid; i < NCAND; i += 256) s[i] = scores[(size_t)b * NCAND + i];
    __syncthreads();

    for (int t = 0; t < KSEL; ++t) {
        float best = -FLT_MAX;
        int   bidx = NCAND;
        for (int i = tid; i < NCAND; i += 256) {
            float v = s[i];
            if (v > best || (v == best && i < bidx)) { best = v; bidx = i; }
        }
        rv[tid] = best; ri[tid] = bidx;
        __syncthreads();
        for (int off = 128; off > 0; off >>= 1) {
            if (tid < off) {
                float vo = rv[tid + off]; int io = ri[tid + off];
                if (vo > rv[tid] || (vo == rv[tid] && io < ri[tid])) {
                    rv[tid] = vo; ri[tid] = io;
                }
            }
            __syncthreads();
        }
        if (tid == 0) { sel[b * KSEL + t] = ri[0]; s[ri[0]] = -FLT_MAX; }
        __syncthreads();
    }
}

// ---------------------------------------------------------------------------
// Kernel 3: gather selected rows -> out[b, j, :]
// ---------------------------------------------------------------------------
__global__ void __launch_bounds__(256)
gather_kernel(const float* __restrict__ value,
              const int* __restrict__ sel,
              float* __restrict__ out)
{
    const int b   = blockIdx.x / KSEL;
    const int j   = blockIdx.x % KSEL;
    const int row = sel[b * KSEL + j];

    const float4* src = (const float4*)(value + ((size_t)b * NCAND + row) * DDIM);
    float4*       dst = (float4*)(out + ((size_t)b * KSEL + j) * DDIM);
    for (int i = threadIdx.x; i < DDIM / 4; i += 256) dst[i] = src[i];
}

// ---------------------------------------------------------------------------
extern "C" void kernel_launch(void* const* d_in, const int* in_sizes, int n_in,
                              void* d_out, int out_size, void* d_ws, size_t ws_size,
                              hipStream_t stream)
{
    (void)in_sizes; (void)n_in; (void)out_size; (void)ws_size;

    const float* value = (const float*)d_in[0];   // [32, 2048, 2048] f32
    const float* w     = (const float*)d_in[1];   // [2048, 1] f32
    const float* bias  = (const float*)d_in[2];   // [1] f32
    // d_in[3] = num_cxt (==5), compiled in as KSEL

    float* scores = (float*)d_ws;                                      // 256 KB
    int*   sel    = (int*)((char*)d_ws + (size_t)BATCH * NCAND * sizeof(float)); // 640 B
    float* out    = (float*)d_out;                                     // [32, 5, 2048]

    score_kernel <<<dim3(BATCH * 16),  dim3(256), 0, stream>>>(value, w, bias, scores);
    topk_kernel  <<<dim3(BATCH),       dim3(256), 0, stream>>>(scores, sel);
    gather_kernel<<<dim3(BATCH * KSEL), dim3(256), 0, stream>>>(value, sel, out);
}